// GraphSAGEVar_29231547416624
// MI455X (gfx1250) — compile-verified
//
#include <hip/hip_runtime.h>

#define N_NODES 100000
#define N_EDGES 1600000
#define F_HID   128
#define F_OUT   40

typedef __attribute__((ext_vector_type(2))) float v2f;
typedef __attribute__((ext_vector_type(8))) float v8f;

// ---------------------------------------------------------------- utilities
__global__ void zero_f(float* __restrict__ p, int n) {
    int i = blockIdx.x * blockDim.x + threadIdx.x;
    if (i < n) p[i] = 0.0f;
}

__global__ void deg_count(const int* __restrict__ dst, float* __restrict__ deg, int E) {
    int e = blockIdx.x * blockDim.x + threadIdx.x;
    if (e < E) atomicAdd(&deg[dst[e]], 1.0f);
}

__global__ void deg_invert(float* __restrict__ deg, int n) {
    int i = blockIdx.x * blockDim.x + threadIdx.x;
    if (i < n) deg[i] = 1.0f / fmaxf(deg[i], 1.0f);
}

// One wave32 per edge: lane l moves features [4l, 4l+3] of the 128-wide row.
// Gather is a coalesced 512B row read; scatter-add resolves in L2.
__global__ void scatter_add(const float* __restrict__ feat, const int* __restrict__ src,
                            const int* __restrict__ dst, float* __restrict__ agg, int E) {
    int e = blockIdx.x * 8 + (threadIdx.x >> 5);
    if (e >= E) return;
    int lane = threadIdx.x & 31;
    int s = src[e], d = dst[e];
    const float4 v = *(const float4*)(feat + (size_t)s * F_HID + lane * 4);
    float* p = agg + (size_t)d * F_HID + lane * 4;
    atomicAdd(p + 0, v.x);
    atomicAdd(p + 1, v.y);
    atomicAdd(p + 2, v.z);
    atomicAdd(p + 3, v.w);
}

__global__ void scale_rows(float* __restrict__ agg, const float* __restrict__ deginv, int total) {
    int i = blockIdx.x * blockDim.x + threadIdx.x;
    if (i < total) agg[i] *= deginv[i >> 7];   // /128 -> node id
}

// ---------------------------------------------------------------- fused WMMA GEMM
// out[i,o] = sum_k A1[i,k]*W1[o,k] (+ sum_k A2[i,k]*W2[o,k]) + bias[o], opt ReLU.
// One wave per 16x16 output tile; blockDim.x = 32 * ceil(OC/16); K fixed at 128.
// A-tiles staged once per block in LDS (shared by all waves); weight rows are
// clamped so the K-loop is branch-free (EXEC all-1s at every WMMA; out-of-range
// columns compute garbage that is never stored).
__global__ __launch_bounds__(256)
void sage_gemm_wmma(const float* __restrict__ A1, const float* __restrict__ W1,
                    const float* __restrict__ A2, const float* __restrict__ W2,
                    const float* __restrict__ bias, float* __restrict__ out,
                    int OC, int relu) {
    __shared__ float smA1[16 * 128];
    __shared__ float smA2[16 * 128];

    const int tid = threadIdx.x;
    const int i0  = blockIdx.x * 16;       // node-row tile

    // cooperative stage of A-tiles: 16 rows x 128 cols, float4 granularity
    for (int idx = tid; idx < 512; idx += blockDim.x) {
        int r  = idx >> 5;                 // 32 float4 per row
        int c4 = (idx & 31) << 2;
        *(float4*)&smA1[r * 128 + c4] = *(const float4*)(A1 + (size_t)(i0 + r) * 128 + c4);
    }
    if (A2) {
        for (int idx = tid; idx < 512; idx += blockDim.x) {
            int r  = idx >> 5;
            int c4 = (idx & 31) << 2;
            *(float4*)&smA2[r * 128 + c4] = *(const float4*)(A2 + (size_t)(i0 + r) * 128 + c4);
        }
    }
    __syncthreads();

    const int wave = tid >> 5;
    const int lane = tid & 31;
    const int o0   = wave * 16;            // output-channel tile
    const int lm   = lane & 15;            // M (A) / N (B,C,D) index for this lane
    const int lk   = (lane >> 4) << 1;     // K pair base: 0 or 2

    const int wrow  = o0 + lm;             // weight row == output column for this lane
    const int wrowc = min(wrow, OC - 1);   // clamped row: keeps loads in-bounds, branch-free

    const float* aL1 = &smA1[lm * 128 + lk];
    const float* w1p = W1 + (size_t)wrowc * 128 + lk;

    v8f c1 = {};
    v8f c2 = {};

#pragma unroll
    for (int k0 = 0; k0 < 128; k0 += 4) {
        v2f a = *(const v2f*)(aL1 + k0);   // ds_load_b64
        v2f b = *(const v2f*)(w1p + k0);   // global_load_b64 (L2-resident weights)
        c1 = __builtin_amdgcn_wmma_f32_16x16x4_f32(false, a, false, b,
                                                   (short)0, c1, false, false);
    }

    if (A2) {
        const float* aL2 = &smA2[lm * 128 + lk];
        const float* w2p = W2 + (size_t)wrowc * 128 + lk;
#pragma unroll
        for (int k0 = 0; k0 < 128; k0 += 4) {
            v2f a = *(const v2f*)(aL2 + k0);
            v2f b = *(const v2f*)(w2p + k0);
            c2 = __builtin_amdgcn_wmma_f32_16x16x4_f32(false, a, false, b,
                                                       (short)0, c2, false, false);
        }
    }

    if (wrow >= OC) return;                // store epilogue only (WMMAs already done)
    const float bv    = bias[wrow];
    const int   rbase = i0 + ((lane >> 4) << 3);   // +0 or +8
#pragma unroll
    for (int r = 0; r < 8; ++r) {
        float v = c1[r] + c2[r] + bv;
        if (relu) v = fmaxf(v, 0.0f);
        out[(size_t)(rbase + r) * OC + wrow] = v;
    }
}

// ---------------------------------------------------------------- softmax / argmax
__global__ void softmax_argmax(const float* __restrict__ logits, float* __restrict__ soft,
                               float* __restrict__ hard, int n) {
    int i = blockIdx.x * blockDim.x + threadIdx.x;
    if (i >= n) return;
    const float* lg = logits + (size_t)i * F_OUT;
    float m = lg[0];
    int   am = 0;
    for (int j = 1; j < F_OUT; ++j) {
        float v = lg[j];
        if (v > m) { m = v; am = j; }      // strict > keeps first max (jnp.argmax)
    }
    float s = 0.0f;
    for (int j = 0; j < F_OUT; ++j) s += expf(lg[j] - m);
    float inv = 1.0f / s;
    for (int j = 0; j < F_OUT; ++j) soft[(size_t)i * F_OUT + j] = expf(lg[j] - m) * inv;
    hard[i] = (float)am;
}

// ---------------------------------------------------------------- launch
extern "C" void kernel_launch(void* const* d_in, const int* in_sizes, int n_in,
                              void* d_out, int out_size, void* d_ws, size_t ws_size,
                              hipStream_t stream) {
    const float* x   = (const float*)d_in[0];
    const int*   ei  = (const int*)d_in[1];
    const int*   src = ei;
    const int*   dst = ei + N_EDGES;
    const float* W1l = (const float*)d_in[2];
    const float* b1  = (const float*)d_in[3];
    const float* W1r = (const float*)d_in[4];
    const float* W2l = (const float*)d_in[5];
    const float* b2  = (const float*)d_in[6];
    const float* W2r = (const float*)d_in[7];
    const float* Wc  = (const float*)d_in[8];
    const float* bc  = (const float*)d_in[9];

    // workspace: agg [N,128] | h1 [N,128] | deg [N]
    float* agg = (float*)d_ws;
    float* h1  = agg + (size_t)N_NODES * F_HID;
    float* deg = h1 + (size_t)N_NODES * F_HID;

    // d_out: logits [N,40] | embedding [N,128] | soft [N,40] | hard [N]
    float* logits = (float*)d_out;
    float* emb    = logits + (size_t)N_NODES * F_OUT;
    float* soft   = emb + (size_t)N_NODES * F_HID;
    float* hard   = soft + (size_t)N_NODES * F_OUT;

    const int NF = N_NODES * F_HID;
    dim3 b256(256);

    // degrees (shared by both layers)
    zero_f<<<(N_NODES + 255) / 256, b256, 0, stream>>>(deg, N_NODES);
    deg_count<<<(N_EDGES + 255) / 256, b256, 0, stream>>>(dst, deg, N_EDGES);
    deg_invert<<<(N_NODES + 255) / 256, b256, 0, stream>>>(deg, N_NODES);

    // ---- layer 1: h1 = relu(mean_agg(x) @ W1l^T + x @ W1r^T + b1)
    zero_f<<<(NF + 255) / 256, b256, 0, stream>>>(agg, NF);
    scatter_add<<<N_EDGES / 8, b256, 0, stream>>>(x, src, dst, agg, N_EDGES);
    scale_rows<<<(NF + 255) / 256, b256, 0, stream>>>(agg, deg, NF);
    sage_gemm_wmma<<<N_NODES / 16, dim3(256), 0, stream>>>(agg, W1l, x, W1r, b1, h1,
                                                           F_HID, /*relu=*/1);

    // ---- layer 2: emb = mean_agg(h1) @ W2l^T + h1 @ W2r^T + b2
    zero_f<<<(NF + 255) / 256, b256, 0, stream>>>(agg, NF);
    scatter_add<<<N_EDGES / 8, b256, 0, stream>>>(h1, src, dst, agg, N_EDGES);
    scale_rows<<<(NF + 255) / 256, b256, 0, stream>>>(agg, deg, NF);
    sage_gemm_wmma<<<N_NODES / 16, dim3(256), 0, stream>>>(agg, W2l, h1, W2r, b2, emb,
                                                           F_HID, /*relu=*/0);

    // ---- classifier: logits = emb @ Wc^T + bc   (40 cols -> 3 o-tiles, 96 threads)
    sage_gemm_wmma<<<N_NODES / 16, dim3(96), 0, stream>>>(emb, Wc, nullptr, nullptr, bc,
                                                          logits, F_OUT, /*relu=*/0);

    // ---- softmax + argmax
    softmax_argmax<<<(N_NODES + 255) / 256, b256, 0, stream>>>(logits, soft, hard, N_NODES);
}